// GCU_25640954757903
// MI455X (gfx1250) — compile-verified
//
#include <hip/hip_runtime.h>
#include <hip/hip_bf16.h>

typedef __attribute__((ext_vector_type(2))) float v2f;
typedef __attribute__((ext_vector_type(8))) float v8f;
typedef __attribute__((ext_vector_type(4))) unsigned int v4u;
typedef __attribute__((ext_vector_type(8))) int v8i;
typedef __attribute__((ext_vector_type(4))) int v4i;

// D = A(16x4 f32) * B(4x16 f32) + C, full f32 WMMA (wave32).
__device__ __forceinline__ v8f wmma4(v2f a, v2f b, v8f c) {
    return __builtin_amdgcn_wmma_f32_16x16x4_f32(
        /*neg_a=*/false, a, /*neg_b=*/false, b,
        /*c_mod=*/(short)0, c, /*reuse_a=*/false, /*reuse_b=*/false);
}

#define D_DIM 2048
#define V_DIM 32
#define N_PIX 4096
#define OUTF  256
#define KSLAB 128
#define MT    2          // 16-row M tiles per wave (32 rows)

// Pair-interleaved B-panel index: element (k, v) -> (k/2)*64 + v*2 + (k&1).
// A lane's WMMA B-fragment {B[kk][v], B[kk+1][v]} is then 8 contiguous bytes.
__device__ __forceinline__ int pidx(int k, int v) {
    return ((k >> 1) << 6) + (v << 1) + (k & 1);
}

// ---------------------------------------------------------------------------
// TDM: async-load a 2D tile (tile_d1 rows x tile_d0 f32 elements, row stride
// stride0 elements) from global to LDS.  D# per CDNA5 ISA ch.8.
__device__ __forceinline__ void tdm_load_2d(
    unsigned lds_off, const void* gptr,
    unsigned tile_d0, unsigned tile_d1, unsigned stride0)
{
    unsigned long long ga = (unsigned long long)(size_t)gptr;
    v4u g0;
    g0[0] = 1u;                                         // count=1, no gather
    g0[1] = lds_off;                                    // lds_addr (bytes)
    g0[2] = (unsigned)(ga & 0xffffffffu);               // global_addr[31:0]
    g0[3] = (unsigned)((ga >> 32) & 0x01ffffffu)        // global_addr[56:32]
          | (2u << 30);                                 // type=2 ("image")
    v8i g1;
    g1[0] = (int)(2u << 16);                            // data_size=2 -> 4 bytes
    g1[1] = (int)((tile_d0 & 0xffffu) << 16);           // tensor_dim0[15:0]
    g1[2] = (int)((tile_d0 >> 16) | ((tile_d1 & 0xffffu) << 16));
    g1[3] = (int)((tile_d1 >> 16) | (tile_d0 << 16));   // td1 hi | tile_dim0
    g1[4] = (int)(tile_d1 & 0xffffu);                   // tile_dim1 | tile_dim2=0
    g1[5] = (int)stride0;                               // tensor_dim0_stride[31:0]
    g1[6] = 0;
    g1[7] = 0;
    v4i z4 = {0, 0, 0, 0};
    v8i z8 = {0, 0, 0, 0, 0, 0, 0, 0};
    __builtin_amdgcn_tensor_load_to_lds(g0, g1, z4, z4, z8, 0);
}

// ---------------------------------------------------------------------------
// K0: B1p = 1/var^2, B2p = -2*W/var^2 (pair-interleaved); c[v] = sum W^2/var^2.
__global__ __launch_bounds__(256) void k_prep(
    const float* __restrict__ W, const float* __restrict__ var,
    float* __restrict__ B1p, float* __restrict__ B2p, float* __restrict__ cvec)
{
    __shared__ float sm[256];
    const int v = blockIdx.x;
    float s = 0.0f;
    for (int k = threadIdx.x; k < D_DIM; k += 256) {
        float w  = W[k * V_DIM + v];
        float vr = var[k * V_DIM + v];
        float inv2 = 1.0f / (vr * vr);
        B1p[pidx(k, v)] = inv2;
        B2p[pidx(k, v)] = -2.0f * w * inv2;
        s += w * w * inv2;
    }
    sm[threadIdx.x] = s; __syncthreads();
    for (int off = 128; off; off >>= 1) {
        if (threadIdx.x < off) sm[threadIdx.x] += sm[threadIdx.x + off];
        __syncthreads();
    }
    if (threadIdx.x == 0) cvec[v] = sm[0];
}

// ---------------------------------------------------------------------------
// K1: Q = (X.^2) @ B1 + X @ B2 + c.  TDM double-buffers the A panel
// (32 x KSLAB, row stride D_DIM) and the pair-interleaved B panels; inner
// loop is contiguous ds_load_b64 fragments + v_wmma.
__global__ __launch_bounds__(32) void k_gemmQ(
    const float* __restrict__ X, const float* __restrict__ B1p,
    const float* __restrict__ B2p, const float* __restrict__ cvec,
    float* __restrict__ Q)
{
    __shared__ __align__(16) float lx [2][MT * 16 * KSLAB];  // A: 32 x KSLAB
    __shared__ __align__(16) float lb1[2][KSLAB * V_DIM];    // B1 (pair-ilv)
    __shared__ __align__(16) float lb2[2][KSLAB * V_DIM];    // B2 (pair-ilv)
    const int lane = threadIdx.x;
    const int half = lane >> 4, lm = lane & 15;
    const int m0 = blockIdx.x * (MT * 16);
    const int NSLAB = D_DIM / KSLAB;          // 16

    auto issue = [&](int s, int buf) {
        tdm_load_2d((unsigned)(size_t)&lx[buf][0],
                    X + (size_t)m0 * D_DIM + (size_t)s * KSLAB,
                    KSLAB, MT * 16, D_DIM);
        tdm_load_2d((unsigned)(size_t)&lb1[buf][0],
                    B1p + (size_t)s * KSLAB * V_DIM, KSLAB * V_DIM, 1, KSLAB * V_DIM);
        tdm_load_2d((unsigned)(size_t)&lb2[buf][0],
                    B2p + (size_t)s * KSLAB * V_DIM, KSLAB * V_DIM, 1, KSLAB * V_DIM);
    };

    v8f acc[MT][2];
    for (int t = 0; t < MT; ++t) { acc[t][0] = (v8f){}; acc[t][1] = (v8f){}; }
    issue(0, 0);
    for (int s = 0; s < NSLAB; ++s) {
        const int buf = s & 1;
        if (s + 1 < NSLAB) {
            issue(s + 1, buf ^ 1);
            __builtin_amdgcn_s_wait_tensorcnt(3);   // slab s (3 ops) done
        } else {
            __builtin_amdgcn_s_wait_tensorcnt(0);
        }
        asm volatile("" ::: "memory");
        for (int k0l = 0; k0l < KSLAB; k0l += 4) {
            const int kk = k0l + half * 2;
            const int pb = ((k0l >> 1) + half) * (2 * V_DIM) + lm * 2;
            v2f b10 = *(const v2f*)&lb1[buf][pb];
            v2f b11 = *(const v2f*)&lb1[buf][pb + 32];
            v2f b20 = *(const v2f*)&lb2[buf][pb];
            v2f b21 = *(const v2f*)&lb2[buf][pb + 32];
            for (int t = 0; t < MT; ++t) {
                v2f a = *(const v2f*)&lx[buf][(t * 16 + lm) * KSLAB + kk];
                v2f axx = a * a;
                acc[t][0] = wmma4(axx, b10, acc[t][0]);
                acc[t][0] = wmma4(a,   b20, acc[t][0]);
                acc[t][1] = wmma4(axx, b11, acc[t][1]);
                acc[t][1] = wmma4(a,   b21, acc[t][1]);
            }
        }
        asm volatile("" ::: "memory");
    }
    const float cv0 = cvec[lm], cv1 = cvec[16 + lm];
    for (int t = 0; t < MT; ++t)
        for (int j = 0; j < 8; ++j) {
            const int row = m0 + t * 16 + j + half * 8;
            Q[(size_t)row * V_DIM + lm]      = acc[t][0][j] + cv0;
            Q[(size_t)row * V_DIM + 16 + lm] = acc[t][1][j] + cv1;
        }
}

// ---------------------------------------------------------------------------
// K2: per-row: Q -= rowmin; Q = exp(-0.5 Q); Q /= rowsum.
// Writes row-major Q (for colsum + final GEMM) and pair-interleaved Qp
// (B-operand of the X^T @ Q GEMM).
__global__ __launch_bounds__(256) void k_rowsm(float* __restrict__ Q,
                                               float* __restrict__ Qp)
{
    const int row = blockIdx.x * blockDim.x + threadIdx.x;
    if (row >= N_PIX) return;
    float q[V_DIM];
    float mn = 3.0e38f;
    for (int v = 0; v < V_DIM; ++v) { q[v] = Q[row * V_DIM + v]; mn = fminf(mn, q[v]); }
    float s = 0.0f;
    for (int v = 0; v < V_DIM; ++v) { q[v] = expf(-0.5f * (q[v] - mn)); s += q[v]; }
    const float inv = 1.0f / s;
    for (int v = 0; v < V_DIM; ++v) {
        const float val = q[v] * inv;
        Q[row * V_DIM + v] = val;
        Qp[pidx(row, v)]   = val;
    }
}

// K3: n[v] = column sum of Q.
__global__ __launch_bounds__(256) void k_colsum(const float* __restrict__ Q,
                                                float* __restrict__ nvec)
{
    __shared__ float sm[256];
    const int v = blockIdx.x;
    float s = 0.0f;
    for (int r = threadIdx.x; r < N_PIX; r += 256) s += Q[r * V_DIM + v];
    sm[threadIdx.x] = s; __syncthreads();
    for (int off = 128; off; off >>= 1) {
        if (threadIdx.x < off) sm[threadIdx.x] += sm[threadIdx.x + off];
        __syncthreads();
    }
    if (threadIdx.x == 0) nvec[v] = sm[0];
}

// ---------------------------------------------------------------------------
// K4: Zpre = X^T @ Q.  TDM stages the X^T panel (KSLAB k-rows x 32 m-cols,
// row stride D_DIM) and the pair-interleaved Qp panel, double buffered.
__global__ __launch_bounds__(32) void k_gemmZ(
    const float* __restrict__ X, const float* __restrict__ Qp,
    float* __restrict__ Zpre)
{
    __shared__ __align__(16) float lx[2][KSLAB * MT * 16]; // X^T: k-major, 32 cols
    __shared__ __align__(16) float lq[2][KSLAB * V_DIM];   // Qp (pair-ilv)
    const int lane = threadIdx.x;
    const int half = lane >> 4, lm = lane & 15;
    const int m0 = blockIdx.x * (MT * 16);
    const int NSLAB = N_PIX / KSLAB;          // 32

    auto issue = [&](int s, int buf) {
        tdm_load_2d((unsigned)(size_t)&lx[buf][0],
                    X + (size_t)s * KSLAB * D_DIM + m0,
                    MT * 16, KSLAB, D_DIM);
        tdm_load_2d((unsigned)(size_t)&lq[buf][0],
                    Qp + (size_t)s * KSLAB * V_DIM, KSLAB * V_DIM, 1, KSLAB * V_DIM);
    };

    v8f acc[MT][2];
    for (int t = 0; t < MT; ++t) { acc[t][0] = (v8f){}; acc[t][1] = (v8f){}; }
    issue(0, 0);
    for (int s = 0; s < NSLAB; ++s) {
        const int buf = s & 1;
        if (s + 1 < NSLAB) {
            issue(s + 1, buf ^ 1);
            __builtin_amdgcn_s_wait_tensorcnt(2);   // slab s (2 ops) done
        } else {
            __builtin_amdgcn_s_wait_tensorcnt(0);
        }
        asm volatile("" ::: "memory");
        for (int k0l = 0; k0l < KSLAB; k0l += 4) {
            const int kk = k0l + half * 2;
            const int pb = ((k0l >> 1) + half) * (2 * V_DIM) + lm * 2;
            v2f b0 = *(const v2f*)&lq[buf][pb];
            v2f b1 = *(const v2f*)&lq[buf][pb + 32];
            for (int t = 0; t < MT; ++t) {
                v2f a;
                a.x = lx[buf][kk       * (MT * 16) + t * 16 + lm];
                a.y = lx[buf][(kk + 1) * (MT * 16) + t * 16 + lm];
                acc[t][0] = wmma4(a, b0, acc[t][0]);
                acc[t][1] = wmma4(a, b1, acc[t][1]);
            }
        }
        asm volatile("" ::: "memory");
    }
    for (int t = 0; t < MT; ++t)
        for (int j = 0; j < 8; ++j) {
            const int m = m0 + t * 16 + j + half * 8;
            Zpre[(size_t)m * V_DIM + lm]      = acc[t][0][j];
            Zpre[(size_t)m * V_DIM + 16 + lm] = acc[t][1][j];
        }
}

// ---------------------------------------------------------------------------
// K5: Z = ((Zpre - W*n)/var)/n (or 1 if n<=0), then Z /= sum_k Z^2 per column.
__global__ __launch_bounds__(256) void k_zfin(
    const float* __restrict__ Zpre, const float* __restrict__ W,
    const float* __restrict__ var, const float* __restrict__ nvec,
    float* __restrict__ Z)
{
    __shared__ float sm[256];
    const int v = blockIdx.x;
    const float nv = nvec[v];
    float vals[8];
    float ss = 0.0f;
    for (int i = 0; i < 8; ++i) {
        const int k = threadIdx.x + i * 256;
        float z = (Zpre[k * V_DIM + v] - W[k * V_DIM + v] * nv) / var[k * V_DIM + v];
        z = (nv > 0.0f) ? (z / nv) : 1.0f;
        vals[i] = z; ss += z * z;
    }
    sm[threadIdx.x] = ss; __syncthreads();
    for (int off = 128; off; off >>= 1) {
        if (threadIdx.x < off) sm[threadIdx.x] += sm[threadIdx.x + off];
        __syncthreads();
    }
    const float inv = 1.0f / sm[0];
    for (int i = 0; i < 8; ++i) {
        const int k = threadIdx.x + i * 256;
        Z[k * V_DIM + v] = vals[i] * inv;
    }
}

// ---------------------------------------------------------------------------
// Generic WMMA GEMM, C = A^T @ B with A stored K-major: A[k*ldA + m].
__global__ __launch_bounds__(32) void k_gemm_tn(
    const float* __restrict__ A, const float* __restrict__ B,
    float* __restrict__ C, int K, int ldA, int ldB, int ldC)
{
    const int lane = threadIdx.x;
    const int half = lane >> 4, lm = lane & 15;
    const int m0 = blockIdx.x * 16, n0 = blockIdx.y * 16;
    v8f acc = {};
    for (int k0 = 0; k0 < K; k0 += 4) {
        const int kk = k0 + half * 2;
        v2f a, b;
        a.x = A[(size_t)kk * ldA + m0 + lm];
        a.y = A[(size_t)(kk + 1) * ldA + m0 + lm];
        b.x = B[(size_t)kk * ldB + n0 + lm];
        b.y = B[(size_t)(kk + 1) * ldB + n0 + lm];
        acc = wmma4(a, b, acc);
    }
    for (int j = 0; j < 8; ++j)
        C[(size_t)(m0 + j + half * 8) * ldC + n0 + lm] = acc[j];
}

// Generic WMMA GEMM, C = A @ B with A row-major: A[m*ldA + k]; optional relu.
__global__ __launch_bounds__(32) void k_gemm_nn(
    const float* __restrict__ A, const float* __restrict__ B,
    float* __restrict__ C, int K, int ldA, int ldB, int ldC, int do_relu)
{
    const int lane = threadIdx.x;
    const int half = lane >> 4, lm = lane & 15;
    const int m0 = blockIdx.x * 16, n0 = blockIdx.y * 16;
    const float* ar = A + (size_t)(m0 + lm) * ldA;
    v8f acc = {};
    for (int k0 = 0; k0 < K; k0 += 4) {
        const int kk = k0 + half * 2;
        v2f a, b;
        a.x = ar[kk]; a.y = ar[kk + 1];
        b.x = B[(size_t)kk * ldB + n0 + lm];
        b.y = B[(size_t)(kk + 1) * ldB + n0 + lm];
        acc = wmma4(a, b, acc);
    }
    for (int j = 0; j < 8; ++j) {
        float val = acc[j];
        if (do_relu) val = fmaxf(val, 0.0f);
        C[(size_t)(m0 + j + half * 8) * ldC + n0 + lm] = val;
    }
}

// ---------------------------------------------------------------------------
extern "C" void kernel_launch(void* const* d_in, const int* in_sizes, int n_in,
                              void* d_out, int out_size, void* d_ws, size_t ws_size,
                              hipStream_t stream)
{
    const float* X    = (const float*)d_in[0];   // (64,64,2048) -> (4096,2048)
    const float* W    = (const float*)d_in[1];   // (2048,32)
    const float* var  = (const float*)d_in[2];   // (2048,32)
    const float* wgt  = (const float*)d_in[3];   // (2048,256)
    float* out = (float*)d_out;                  // (1,256,64,64) raw == (4096,256)

    float* ws   = (float*)d_ws;
    float* B1p  = ws;               ws += D_DIM * V_DIM;
    float* B2p  = ws;               ws += D_DIM * V_DIM;
    float* cvec = ws;               ws += 64;
    float* Q    = ws;               ws += N_PIX * V_DIM;
    float* Qp   = ws;               ws += N_PIX * V_DIM;
    float* nvec = ws;               ws += 64;
    float* Zpre = ws;               ws += D_DIM * V_DIM;
    float* Z    = ws;               ws += D_DIM * V_DIM;
    float* T    = ws;               ws += V_DIM * OUTF;
    float* Adj  = ws;               ws += V_DIM * V_DIM;
    float* outb = ws;               ws += V_DIM * OUTF;

    k_prep<<<dim3(V_DIM), dim3(256), 0, stream>>>(W, var, B1p, B2p, cvec);
    k_gemmQ<<<dim3(N_PIX / (MT * 16)), dim3(32), 0, stream>>>(X, B1p, B2p, cvec, Q);
    k_rowsm<<<dim3(N_PIX / 256), dim3(256), 0, stream>>>(Q, Qp);
    k_colsum<<<dim3(V_DIM), dim3(256), 0, stream>>>(Q, nvec);
    k_gemmZ<<<dim3(D_DIM / (MT * 16)), dim3(32), 0, stream>>>(X, Qp, Zpre);
    k_zfin<<<dim3(V_DIM), dim3(256), 0, stream>>>(Zpre, W, var, nvec, Z);
    k_gemm_tn<<<dim3(V_DIM / 16, OUTF / 16), dim3(32), 0, stream>>>(
        Z, wgt, T, D_DIM, V_DIM, OUTF, OUTF);
    k_gemm_tn<<<dim3(V_DIM / 16, V_DIM / 16), dim3(32), 0, stream>>>(
        Z, Z, Adj, D_DIM, V_DIM, V_DIM, V_DIM);
    k_gemm_nn<<<dim3(V_DIM / 16, OUTF / 16), dim3(32), 0, stream>>>(
        Adj, T, outb, V_DIM, V_DIM, OUTF, OUTF, 1);
    k_gemm_nn<<<dim3(N_PIX / 16, OUTF / 16), dim3(32), 0, stream>>>(
        Q, outb, out, V_DIM, V_DIM, OUTF, OUTF, 0);
}